// intl_43765716746255
// MI455X (gfx1250) — compile-verified
//
#include <hip/hip_runtime.h>

typedef __attribute__((ext_vector_type(2))) float v2f;
typedef __attribute__((ext_vector_type(8))) float v8f;

#define WMMA_F32X4(a, b, c) \
  __builtin_amdgcn_wmma_f32_16x16x4_f32(false, (a), false, (b), (short)0, (c), false, false)

constexpr int NB = 256;   // batch
constexpr int MX = 512;   // rows of x per batch
constexpr int D  = 256;   // feature dim
constexpr int KB = 32;    // K block staged in LDS
constexpr int TM = 64;    // output tile rows per block
constexpr int TN = 128;   // output tile cols per block (8 waves, 32x32 strip each)

// ---------------- mean over M axis ----------------
__global__ void mean_kernel(const float* __restrict__ x, float* __restrict__ mean) {
  const int n = blockIdx.x;
  const int d = threadIdx.x;                       // 256 threads = D
  const float* xp = x + (size_t)n * MX * D + d;
  float s = 0.0f;
  for (int m = 0; m < MX; ++m) s += xp[(size_t)m * D];
  mean[n * D + d] = s * (1.0f / MX);
}

// ---------------- P = I ----------------
__global__ void eye_kernel(float* __restrict__ P) {
  const int n = blockIdx.y;
  const int i = blockIdx.x;
  const int j = threadIdx.x;
  P[((size_t)n * D + i) * D + j] = (i == j) ? 1.0f : 0.0f;
}

// ---------------- trace of sigma ----------------
__global__ void trace_kernel(const float* __restrict__ sigma, float* __restrict__ trace) {
  __shared__ float red[256];
  const int n = blockIdx.x;
  const int t = threadIdx.x;
  red[t] = sigma[(size_t)n * D * D + (size_t)t * D + t];
  __syncthreads();
  for (int s = 128; s > 0; s >>= 1) {
    if (t < s) red[t] += red[t + s];
    __syncthreads();
  }
  if (t == 0) trace[n] = red[0];
}

// ---------------- batched WMMA f32 GEMM ----------------
// OP == 0:  C[n,D,D]  = s * A@B + beta*Cadd          (A,B: [n,D,D])
// OP == 1:  C[n,D,D]  = s * (x-mean)^T (x-mean)      (A=B=x: [n,MX,D], K=MX)
// OP == 2:  C[n,MX,D] = s * (x-mean) @ B             (A=x: [n,MX,D], B: [n,D,D])
// traceMode: 0 -> s=alpha ; 1 -> s=alpha/trace[n] ; 2 -> s=alpha*rsqrt(trace[n])
// All matrices have leading dimension D (=256).
template <int OP>
__global__ __launch_bounds__(256) void gemm_kernel(
    const float* __restrict__ A, const float* __restrict__ B,
    const float* __restrict__ Cadd, float* __restrict__ C,
    const float* __restrict__ mean, const float* __restrict__ trace,
    float alpha, float beta, int traceMode) {
  constexpr int Kdim = (OP == 1) ? MX : D;
  constexpr int tilesN = D / TN;                         // 2
  constexpr size_t strA = (OP == 0) ? (size_t)D * D : (size_t)MX * D;
  constexpr size_t strB = (OP == 1) ? (size_t)MX * D : (size_t)D * D;
  constexpr size_t strC = (OP == 2) ? (size_t)MX * D : (size_t)D * D;

  const int n  = blockIdx.y;
  const int i0 = (blockIdx.x / tilesN) * TM;
  const int j0 = (blockIdx.x % tilesN) * TN;
  const int tid = threadIdx.x;                           // 256 threads = 8 wave32

  // A: row-padded to even stride -> aligned ds_load_b64 fragments, conflict-free spread
  __shared__ float As[2][TM][KB + 2];
  // B: pre-interleaved K-pairs -> fragment {B[k][j],B[k+1][j]} is ONE ds_load_b64
  __shared__ v2f Bs[2][KB / 2][TN];

  const float* Ab = A + (size_t)n * strA;
  const float* Bb = B + (size_t)n * strB;

  const int lane = tid & 31;
  const int wid  = tid >> 5;          // 0..7
  const int wr   = wid >> 2;          // 0..1 -> M offset 32*wr
  const int wc   = wid & 3;           // 0..3 -> N offset 32*wc
  const int half = lane >> 4;         // K pair {0,1} vs {2,3}
  const int ln   = lane & 15;

  v8f acc[2][2] = {};

  float ra[8];                        // staged A: 8 floats/thread
  v2f   rb[8];                        // staged B: 8 k-pairs/thread

  auto stage_load = [&](int k0) {
#pragma unroll
    for (int r = 0; r < 8; ++r) {
      const int idx = tid + 256 * r;
      if (OP == 1) {                  // transposed + mean-centered
        const int k = idx >> 6, i = idx & 63;
        ra[r] = Ab[(size_t)(k0 + k) * D + (i0 + i)] - mean[n * D + i0 + i];
      } else if (OP == 2) {           // mean-centered along K
        const int i = idx >> 5, k = idx & 31;
        ra[r] = Ab[(size_t)(i0 + i) * D + (k0 + k)] - mean[n * D + k0 + k];
      } else {
        const int i = idx >> 5, k = idx & 31;
        ra[r] = Ab[(size_t)(i0 + i) * D + (k0 + k)];
      }
      const int kp = idx >> 7, j = idx & 127;
      float b0 = Bb[(size_t)(k0 + 2 * kp + 0) * D + (j0 + j)];
      float b1 = Bb[(size_t)(k0 + 2 * kp + 1) * D + (j0 + j)];
      if (OP == 1) {
        const float mj = mean[n * D + j0 + j];
        b0 -= mj; b1 -= mj;
      }
      rb[r].x = b0; rb[r].y = b1;
    }
  };
  auto stage_store = [&](int p) {
#pragma unroll
    for (int r = 0; r < 8; ++r) {
      const int idx = tid + 256 * r;
      if (OP == 1) { const int k = idx >> 6, i = idx & 63; As[p][i][k] = ra[r]; }
      else         { const int i = idx >> 5, k = idx & 31; As[p][i][k] = ra[r]; }
      const int kp = idx >> 7, j = idx & 127;
      Bs[p][kp][j] = rb[r];
    }
  };

  constexpr int nKB = Kdim / KB;
  stage_load(0);
  stage_store(0);
  __syncthreads();

  for (int ib = 0; ib < nKB; ++ib) {
    const int p = ib & 1;
    const bool hasNext = (ib + 1 < nKB);
    if (hasNext) stage_load((ib + 1) * KB);          // global loads overlap WMMA below
    if (ib + 2 < nKB) {                              // L2 lookahead (global_prefetch_b8)
      const int kq = (ib + 2) * KB;
      __builtin_prefetch(&Ab[(OP == 1) ? (size_t)(kq + (tid >> 6)) * D + i0 + (tid & 63)
                                       : (size_t)(i0 + (tid >> 5)) * D + kq + (tid & 31)], 0, 1);
      __builtin_prefetch(&Bb[(size_t)(kq + 2 * (tid >> 7)) * D + j0 + (tid & 127)], 0, 1);
    }

#pragma unroll
    for (int kk = 0; kk < KB; kk += 4) {
      const int kf = kk + 2 * half;
      const v2f a0 = *(const v2f*)&As[p][32 * wr + ln][kf];
      const v2f a1 = *(const v2f*)&As[p][32 * wr + 16 + ln][kf];
      const v2f b0 = Bs[p][kf >> 1][32 * wc + ln];
      const v2f b1 = Bs[p][kf >> 1][32 * wc + 16 + ln];
      acc[0][0] = WMMA_F32X4(a0, b0, acc[0][0]);
      acc[0][1] = WMMA_F32X4(a0, b1, acc[0][1]);
      acc[1][0] = WMMA_F32X4(a1, b0, acc[1][0]);
      acc[1][1] = WMMA_F32X4(a1, b1, acc[1][1]);
    }

    if (hasNext) stage_store(p ^ 1);                 // DS in-order: after compute's loads
    __syncthreads();
  }

  float s = alpha;
  if (traceMode == 1)      s *= 1.0f / trace[n];
  else if (traceMode == 2) s *= rsqrtf(trace[n]);

  float* Cb = C + (size_t)n * strC;
  const float* Zb = Cadd ? (Cadd + (size_t)n * strC) : nullptr;
#pragma unroll
  for (int sm = 0; sm < 2; ++sm)
#pragma unroll
    for (int sn = 0; sn < 2; ++sn)
#pragma unroll
      for (int r = 0; r < 8; ++r) {
        const int row = i0 + 32 * wr + 16 * sm + r + 8 * half;  // C: vgpr r -> M = r+8*half
        const int col = j0 + 32 * wc + 16 * sn + ln;
        const size_t o = (size_t)row * D + col;
        float v = s * acc[sm][sn][r];
        if (Zb) v += beta * Zb[o];
        Cb[o] = v;
      }
}

extern "C" void kernel_launch(void* const* d_in, const int* in_sizes, int n_in,
                              void* d_out, int out_size, void* d_ws, size_t ws_size,
                              hipStream_t stream) {
  const float* x = (const float*)d_in[0];
  float* out = (float*)d_out;
  float* ws  = (float*)d_ws;

  const size_t matElems = (size_t)NB * D * D;            // 16,777,216 floats
  float* mean  = ws;                                     // NB*D
  float* trace = ws + (size_t)NB * D;                    // NB (padded to 512)
  float* sigma = trace + 512;
  float* P     = sigma + matElems;
  float* T1    = P + matElems;
  float* T2    = T1 + matElems;

  // 1) mean over M
  mean_kernel<<<NB, 256, 0, stream>>>(x, mean);

  // 2) sigma = (x-mean)^T (x-mean) / (M-1)   [not yet trace-normalized]
  gemm_kernel<1><<<dim3((D / TM) * (D / TN), NB), 256, 0, stream>>>(
      x, x, nullptr, sigma, mean, nullptr, 1.0f / (float)(MX - 1), 0.0f, 0);

  // 3) trace(sigma)
  trace_kernel<<<NB, 256, 0, stream>>>(sigma, trace);

  // 4) P = I
  eye_kernel<<<dim3(D, NB), D, 0, stream>>>(P);

  // 5) Newton-Schulz: P <- 1.5 P - 0.5 (P P P) (sigma/tr)
  float* Pc = P;
  float* tA = T1;
  float* tB = T2;
  const dim3 gNS((D / TM) * (D / TN), NB);
  for (int it = 0; it < 4; ++it) {
    gemm_kernel<0><<<gNS, 256, 0, stream>>>(Pc, Pc, nullptr, tA, nullptr, nullptr, 1.0f, 0.0f, 0);
    gemm_kernel<0><<<gNS, 256, 0, stream>>>(tA, Pc, nullptr, tB, nullptr, nullptr, 1.0f, 0.0f, 0);
    gemm_kernel<0><<<gNS, 256, 0, stream>>>(tB, sigma, Pc, tA, nullptr, trace, -0.5f, 1.5f, 1);
    float* t = Pc; Pc = tA; tA = t;                      // P <- new P
  }

  // 6) out = (x - mean) @ P * rsqrt(trace)
  gemm_kernel<2><<<dim3((MX / TM) * (D / TN), NB), 256, 0, stream>>>(
      x, Pc, nullptr, out, mean, trace, 1.0f, 0.0f, 2);
}